// GCN_833223655738
// MI455X (gfx1250) — compile-verified
//
#include <hip/hip_runtime.h>
#include <hip/hip_bf16.h>

// ---------------------------------------------------------------------------
// GNN: 3x GraphConv(64) + BN/ReLU + mean-pool + linear head, for MI455X.
// Scatter-bound workload; GEMMs done with V_WMMA_F32_16X16X4_F32 (full fp32).
// ---------------------------------------------------------------------------

#define NN   50000
#define EE   800000
#define GG   256
#define HID  64
#define EPSF 1e-5f

typedef __attribute__((ext_vector_type(2))) float v2f;
typedef __attribute__((ext_vector_type(8))) float v8f;

// ---------------- zero fill ----------------
__global__ void k_zero(float* __restrict__ p, int n) {
    int i = blockIdx.x * blockDim.x + threadIdx.x;
    if (i < n) p[i] = 0.0f;
}

// ---------------- edge gather * w -> atomic scatter-add ----------------
// 16 threads per edge, float4 per thread (64 feats).
__global__ void k_scatter(const float* __restrict__ feat,
                          const int* __restrict__ ei,   // [2,E] row-major
                          const float* __restrict__ ew,
                          float* __restrict__ agg) {
    int t = blockIdx.x * blockDim.x + threadIdx.x;
    if (t >= EE * 16) return;
    int e  = t >> 4;
    int f4 = (t & 15) * 4;
    int src = ei[e];
    int dst = ei[EE + e];
    float w = ew[e];
    const float4 v = *(const float4*)(feat + (long long)src * HID + f4);
    float* d = agg + (long long)dst * HID + f4;
    atomicAdd(d + 0, w * v.x);
    atomicAdd(d + 1, w * v.y);
    atomicAdd(d + 2, w * v.z);
    atomicAdd(d + 3, w * v.w);
}

// ---------------- H = X@Ws + A@Wn + b  (fp32 WMMA 16x16x4) ----------------
// One wave per 16-row tile. 4 column tiles of 16. K stepped by 4.
// A-matrix (16x4 f32): lane holds row=lane&15, K = (lane>>4)*2 + {0,1}.
// C/D (16x16 f32): vgpr r -> M = r + (lane>>4)*8, N = lane&15.
__global__ __launch_bounds__(256)
void k_gemm(const float* __restrict__ X,
            const float* __restrict__ A,
            const float* __restrict__ Ws,
            const float* __restrict__ Wn,
            const float* __restrict__ bias,
            float* __restrict__ H) {
    const int wave = threadIdx.x >> 5;              // 0..7
    const int lane = threadIdx.x & 31;
    const int row0 = (blockIdx.x * 8 + wave) * 16;  // wave-uniform
    if (row0 >= NN) return;                          // whole wave exits: EXEC stays all-1s
    const int rr = lane & 15;                        // A row / B,C col within tile
    const int kk = (lane >> 4) * 2;                  // K sub-offset (0 or 2)
    const int mh = (lane >> 4) * 8;                  // C row sub-offset (0 or 8)

    v8f c0 = {}, c1 = {}, c2 = {}, c3 = {};

    for (int k = 0; k < HID; k += 4) {
        v2f a = *(const v2f*)(X + (long long)(row0 + rr) * HID + k + kk);
        const float* w0 = Ws + (k + kk) * HID + rr;
        v2f b0, b1, b2, b3;
        b0.x = w0[0];        b0.y = w0[HID];
        b1.x = w0[16];       b1.y = w0[HID + 16];
        b2.x = w0[32];       b2.y = w0[HID + 32];
        b3.x = w0[48];       b3.y = w0[HID + 48];
        c0 = __builtin_amdgcn_wmma_f32_16x16x4_f32(false, a, false, b0, (short)0, c0, false, false);
        c1 = __builtin_amdgcn_wmma_f32_16x16x4_f32(false, a, false, b1, (short)0, c1, false, false);
        c2 = __builtin_amdgcn_wmma_f32_16x16x4_f32(false, a, false, b2, (short)0, c2, false, false);
        c3 = __builtin_amdgcn_wmma_f32_16x16x4_f32(false, a, false, b3, (short)0, c3, false, false);
    }
    for (int k = 0; k < HID; k += 4) {
        v2f a = *(const v2f*)(A + (long long)(row0 + rr) * HID + k + kk);
        const float* w0 = Wn + (k + kk) * HID + rr;
        v2f b0, b1, b2, b3;
        b0.x = w0[0];        b0.y = w0[HID];
        b1.x = w0[16];       b1.y = w0[HID + 16];
        b2.x = w0[32];       b2.y = w0[HID + 32];
        b3.x = w0[48];       b3.y = w0[HID + 48];
        c0 = __builtin_amdgcn_wmma_f32_16x16x4_f32(false, a, false, b0, (short)0, c0, false, false);
        c1 = __builtin_amdgcn_wmma_f32_16x16x4_f32(false, a, false, b1, (short)0, c1, false, false);
        c2 = __builtin_amdgcn_wmma_f32_16x16x4_f32(false, a, false, b2, (short)0, c2, false, false);
        c3 = __builtin_amdgcn_wmma_f32_16x16x4_f32(false, a, false, b3, (short)0, c3, false, false);
    }

    float bi0 = bias[rr];
    float bi1 = bias[16 + rr];
    float bi2 = bias[32 + rr];
    float bi3 = bias[48 + rr];
#pragma unroll
    for (int r = 0; r < 8; ++r) {
        long long row = row0 + r + mh;
        float* hr = H + row * HID;
        hr[rr]      = c0[r] + bi0;
        hr[16 + rr] = c1[r] + bi1;
        hr[32 + rr] = c2[r] + bi2;
        hr[48 + rr] = c3[r] + bi3;
    }
}

// ---------------- per-column sum / sumsq (for batchnorm) ----------------
// stats[0:64]=sum, stats[64:128]=sumsq (pre-zeroed), [128:192]=scale, [192:256]=shift
__global__ __launch_bounds__(256)
void k_colstats(const float* __restrict__ H, float* __restrict__ stats) {
    int col = threadIdx.x & 63;
    int sub = threadIdx.x >> 6;   // 0..3
    float s = 0.f, q = 0.f;
    for (int row = blockIdx.x * 4 + sub; row < NN; row += gridDim.x * 4) {
        float v = H[(long long)row * HID + col];
        s += v; q += v * v;
    }
    __shared__ float ls[256], lq[256];
    ls[threadIdx.x] = s;
    lq[threadIdx.x] = q;
    __syncthreads();
    if (sub == 0) {
        s = ls[col] + ls[64 + col] + ls[128 + col] + ls[192 + col];
        q = lq[col] + lq[64 + col] + lq[128 + col] + lq[192 + col];
        atomicAdd(&stats[col], s);
        atomicAdd(&stats[64 + col], q);
    }
}

__global__ void k_bn_finalize(float* __restrict__ stats,
                              const float* __restrict__ gamma,
                              const float* __restrict__ beta) {
    int i = threadIdx.x;  // 64 threads
    float mean = stats[i] / (float)NN;
    float var  = stats[64 + i] / (float)NN - mean * mean;
    float sc   = gamma[i] * rsqrtf(var + EPSF);
    stats[128 + i] = sc;
    stats[192 + i] = beta[i] - mean * sc;
}

// ---------------- in-place affine + relu ----------------
__global__ void k_bnrelu(float* __restrict__ H, const float* __restrict__ stats) {
    int t = blockIdx.x * blockDim.x + threadIdx.x;
    if (t >= NN * 16) return;
    int c4 = (t & 15) * 4;
    float4 v = *(float4*)(H + (long long)t * 4);
    v.x = fmaxf(v.x * stats[128 + c4 + 0] + stats[192 + c4 + 0], 0.f);
    v.y = fmaxf(v.y * stats[128 + c4 + 1] + stats[192 + c4 + 1], 0.f);
    v.z = fmaxf(v.z * stats[128 + c4 + 2] + stats[192 + c4 + 2], 0.f);
    v.w = fmaxf(v.w * stats[128 + c4 + 3] + stats[192 + c4 + 3], 0.f);
    *(float4*)(H + (long long)t * 4) = v;
}

// ---------------- segment-sum pool (atomics) ----------------
__global__ void k_pool(const float* __restrict__ H, const int* __restrict__ batch,
                       float* __restrict__ psum, float* __restrict__ pcnt) {
    int t = blockIdx.x * blockDim.x + threadIdx.x;
    if (t >= NN * 16) return;
    int n  = t >> 4;
    int f4 = (t & 15) * 4;
    int g  = batch[n];
    float4 v = *(const float4*)(H + (long long)n * HID + f4);
    float* d = psum + (long long)g * HID + f4;
    atomicAdd(d + 0, v.x);
    atomicAdd(d + 1, v.y);
    atomicAdd(d + 2, v.z);
    atomicAdd(d + 3, v.w);
    if ((t & 15) == 0) atomicAdd(&pcnt[g], 1.0f);
}

// ---------------- mean + linear head ----------------
__global__ void k_head(const float* __restrict__ psum, const float* __restrict__ pcnt,
                       const float* __restrict__ Wlin, const float* __restrict__ blin,
                       float* __restrict__ out) {
    int g = blockIdx.x;
    int j = threadIdx.x;  // 64 threads
    float c = fmaxf(pcnt[g], 1.0f);
    float p = psum[(long long)g * HID + j] / c;
    __shared__ float a0[64], a1[64];
    a0[j] = p * Wlin[j * 2 + 0];
    a1[j] = p * Wlin[j * 2 + 1];
    __syncthreads();
    for (int s = 32; s > 0; s >>= 1) {
        if (j < s) { a0[j] += a0[j + s]; a1[j] += a1[j + s]; }
        __syncthreads();
    }
    if (j == 0) {
        out[g * 2 + 0] = a0[0] + blin[0];
        out[g * 2 + 1] = a1[0] + blin[1];
    }
}

// ---------------------------------------------------------------------------
extern "C" void kernel_launch(void* const* d_in, const int* in_sizes, int n_in,
                              void* d_out, int out_size, void* d_ws, size_t ws_size,
                              hipStream_t stream) {
    const float* x     = (const float*)d_in[0];
    const int*   ei    = (const int*)d_in[1];
    const float* ew    = (const float*)d_in[2];
    const int*   batch = (const int*)d_in[3];
    const float* W1s = (const float*)d_in[4];
    const float* W1n = (const float*)d_in[5];
    const float* b1  = (const float*)d_in[6];
    const float* W2s = (const float*)d_in[7];
    const float* W2n = (const float*)d_in[8];
    const float* b2  = (const float*)d_in[9];
    const float* W3s = (const float*)d_in[10];
    const float* W3n = (const float*)d_in[11];
    const float* b3  = (const float*)d_in[12];
    const float* g1  = (const float*)d_in[13];
    const float* be1 = (const float*)d_in[14];
    const float* g2  = (const float*)d_in[15];
    const float* be2 = (const float*)d_in[16];
    const float* Wlin = (const float*)d_in[17];
    const float* blin = (const float*)d_in[18];
    float* out = (float*)d_out;

    char* ws = (char*)d_ws;
    const size_t HBYTES = (size_t)NN * HID * sizeof(float);  // 12.8 MB
    float* agg   = (float*)(ws);
    float* h1    = (float*)(ws + HBYTES);
    float* h2    = (float*)(ws + 2 * HBYTES);
    float* h3    = h1;  // h1 dead by the time layer 3 writes
    float* stats = (float*)(ws + 3 * HBYTES);                 // 256 floats
    float* psum  = (float*)(ws + 3 * HBYTES + 1024);          // G*64 floats
    float* pcnt  = psum + GG * HID;                           // G floats (contiguous)

    const int feat_elems = NN * HID;                // 3.2M
    const int scat_thr   = EE * 16;                 // 12.8M
    const int vec_thr    = NN * 16;                 // 800k
    const int gemm_grid  = (NN / 16 + 7) / 8;       // 391

    // ---- layer 1 ----
    k_zero<<<(feat_elems + 255) / 256, 256, 0, stream>>>(agg, feat_elems);
    k_scatter<<<(scat_thr + 255) / 256, 256, 0, stream>>>(x, ei, ew, agg);
    k_gemm<<<gemm_grid, 256, 0, stream>>>(x, agg, W1s, W1n, b1, h1);
    k_zero<<<1, 256, 0, stream>>>(stats, 128);
    k_colstats<<<256, 256, 0, stream>>>(h1, stats);
    k_bn_finalize<<<1, 64, 0, stream>>>(stats, g1, be1);
    k_bnrelu<<<(vec_thr + 255) / 256, 256, 0, stream>>>(h1, stats);

    // ---- layer 2 ----
    k_zero<<<(feat_elems + 255) / 256, 256, 0, stream>>>(agg, feat_elems);
    k_scatter<<<(scat_thr + 255) / 256, 256, 0, stream>>>(h1, ei, ew, agg);
    k_gemm<<<gemm_grid, 256, 0, stream>>>(h1, agg, W2s, W2n, b2, h2);
    k_zero<<<1, 256, 0, stream>>>(stats, 128);
    k_colstats<<<256, 256, 0, stream>>>(h2, stats);
    k_bn_finalize<<<1, 64, 0, stream>>>(stats, g2, be2);
    k_bnrelu<<<(vec_thr + 255) / 256, 256, 0, stream>>>(h2, stats);

    // ---- layer 3 (no BN) ----
    k_zero<<<(feat_elems + 255) / 256, 256, 0, stream>>>(agg, feat_elems);
    k_scatter<<<(scat_thr + 255) / 256, 256, 0, stream>>>(h2, ei, ew, agg);
    k_gemm<<<gemm_grid, 256, 0, stream>>>(h2, agg, W3s, W3n, b3, h3);

    // ---- pool + head ----
    k_zero<<<(GG * (HID + 1) + 255) / 256, 256, 0, stream>>>(psum, GG * (HID + 1));
    k_pool<<<(vec_thr + 255) / 256, 256, 0, stream>>>(h3, batch, psum, pcnt);
    k_head<<<GG, 64, 0, stream>>>(psum, pcnt, Wlin, blin, out);
}